// HierarchicalPoseEncoder_58514634440790
// MI455X (gfx1250) — compile-verified
//
#include <hip/hip_runtime.h>
#include <hip/hip_bf16.h>
#include <math.h>

typedef __attribute__((ext_vector_type(2))) float v2f;
typedef __attribute__((ext_vector_type(8))) float v8f;

#define NJ 24
#define THREADS 32
// per 16-element tile: featbuf 24*16*10 = 3840, scratch 16*12 = 192, hbuf 16*20 = 320
#define FB_FLOATS 3840
#define SC_FLOATS 192
#define HB_FLOATS 320
#define TILE_LDS (FB_FLOATS + SC_FLOATS + HB_FLOATS)   // 4352 floats
// one wave per block, two 16-row M-tiles per wave -> 2 * TILE_LDS = 34816 B

__constant__ int c_par[NJ] = {-1,0,0,0,1,2,3,4,5,6,7,8,9,9,9,12,13,14,16,17,18,19,20,21};
// up to 3 children per joint, -1 = none
__constant__ int c_child[NJ * 3] = {
   1,  2,  3,   // 0
   4, -1, -1,   // 1
   5, -1, -1,   // 2
   6, -1, -1,   // 3
   7, -1, -1,   // 4
   8, -1, -1,   // 5
   9, -1, -1,   // 6
  10, -1, -1,   // 7
  11, -1, -1,   // 8
  12, 13, 14,   // 9
  -1, -1, -1,   // 10
  -1, -1, -1,   // 11
  15, -1, -1,   // 12
  16, -1, -1,   // 13
  17, -1, -1,   // 14
  -1, -1, -1,   // 15
  18, -1, -1,   // 16
  19, -1, -1,   // 17
  20, -1, -1,   // 18
  21, -1, -1,   // 19
  22, -1, -1,   // 20
  23, -1, -1,   // 21
  -1, -1, -1,   // 22
  -1, -1, -1    // 23
};

static __device__ __forceinline__ v8f wmma_f32(v2f a, v2f b, v8f c) {
  // D(16x16,f32) = A(16x4,f32) x B(4x16,f32) + C
  return __builtin_amdgcn_wmma_f32_16x16x4_f32(false, a, false, b, (short)0, c,
                                               false, false);
}

__global__ __launch_bounds__(THREADS)
void pose_encoder_kernel(const float* __restrict__ rots,
                         const float* __restrict__ Jtrs,
                         const float* __restrict__ W0p,
                         const float* __restrict__ b0p,
                         const float* __restrict__ W1p,
                         const float* __restrict__ b1p,
                         const float* __restrict__ W2p,
                         const float* __restrict__ b2p,
                         float* __restrict__ outp) {
  __shared__ float lds[2 * TILE_LDS];

  const int lane = threadIdx.x & 31;
  const int mA   = lane & 15;   // row within a 16-tile / B-column index
  const int n    = lane & 15;
  const int mh   = lane >> 4;   // half-wave (K-pair / row-half select)

  const size_t base = (size_t)blockIdx.x * 32;   // 32 batch elements per wave

  float* const FB0 = lds;                    // tile0 featbuf [j*160 + m*10 + rel]
  float* const SC0 = FB0 + FB_FLOATS;        // tile0 scratch [m*12 + s]
  float* const HB0 = SC0 + SC_FLOATS;        // tile0 hbuf    [m*20 + s]
  float* const FB1 = lds + TILE_LDS;
  float* const SC1 = FB1 + FB_FLOATS;
  float* const HB1 = SC1 + SC_FLOATS;

  // per-lane "my element" pointers: lane e handles batch element base+e
  float* const FBl = (lane < 16) ? FB0 : FB1;
  float* const SCl = (lane < 16) ? SC0 : SC1;
  float* const HBl = (lane < 16) ? HB0 : HB1;
  const float* const myrot = rots + (base + lane) * 216;
  const float* const myjtr = Jtrs + (base + lane) * 72;

  // --- zero K-padding slots (k=19) for both tiles; all 32 lanes active ---
  HBl[mA * 20 + 19] = 0.f;
  #pragma unroll
  for (int j = 0; j < NJ; ++j)
    FBl[j * 160 + mA * 10 + 7] = 0.f;        // rel 7 <-> k = 19

  // ================= Phase 1: global_feat = [rots|Jtrs] @ W0^T + b0 =========
  const float nsel6 = (n < 6) ? 1.f : 0.f;
  v8f ga, gb;
  {
    const float bv = (n < 6) ? b0p[n] : 0.f;
    #pragma unroll
    for (int r = 0; r < 8; ++r) { ga[r] = bv; gb[r] = bv; }
  }
  {
    const float* arot0 = rots + (base + mA) * 216;
    const float* arot1 = arot0 + 16 * 216;
    const float* ajtr0 = Jtrs + (base + mA) * 72;
    const float* ajtr1 = ajtr0 + 16 * 72;
    const float* w0row = W0p + (size_t)((n < 6) ? n : 0) * 288;
    #pragma unroll 2
    for (int kc = 0; kc < 216; kc += 4) {
      const int k = kc + mh * 2;
      v2f bb; bb.x = w0row[k] * nsel6; bb.y = w0row[k + 1] * nsel6;
      v2f a0 = *(const v2f*)(arot0 + k);     // 8B-aligned global b64
      v2f a1 = *(const v2f*)(arot1 + k);
      ga = wmma_f32(a0, bb, ga);             // two independent chains
      gb = wmma_f32(a1, bb, gb);
    }
    #pragma unroll 2
    for (int kc = 0; kc < 72; kc += 4) {
      const int k = kc + mh * 2;
      v2f bb; bb.x = w0row[216 + k] * nsel6; bb.y = w0row[216 + k + 1] * nsel6;
      v2f a0 = *(const v2f*)(ajtr0 + k);
      v2f a1 = *(const v2f*)(ajtr1 + k);
      ga = wmma_f32(a0, bb, ga);
      gb = wmma_f32(a1, bb, gb);
    }
  }
  // root's feat = global_feat -> featbuf[0], rel 1..6 (k = 13..18)
  if (n < 6) {
    #pragma unroll
    for (int r = 0; r < 8; ++r) {
      FB0[(r + 8 * mh) * 10 + 1 + n] = ga[r];
      FB1[(r + 8 * mh) * 10 + 1 + n] = gb[r];
    }
  }

  // ================= Phase 2: kinematic-tree MLP chain ======================
  for (int j = 0; j < NJ; ++j) {
    const int p = c_par[j];
    const float* w1j = W1p + (size_t)j * 361;   // 19x19 row-major
    const float* w2j = W2p + (size_t)j * 114;   // 6x19  row-major

    // ---- build in_feat A-buffers; every lane builds its own element ----
    {
      const float* rsrc = myrot + (size_t)j * 9;
      #pragma unroll
      for (int i = 0; i < 9; ++i) SCl[mA * 12 + i] = rsrc[i];
      const float* jsrc = myjtr + (size_t)j * 3;
      const float x = jsrc[0], y = jsrc[1], z = jsrc[2];
      SCl[mA * 12 + 9]  = x;
      SCl[mA * 12 + 10] = y;
      SCl[mA * 12 + 11] = z;
      float dx = x, dy = y, dz = z;
      if (p >= 0) {
        const float* psrc = myjtr + (size_t)p * 3;
        dx -= psrc[0]; dy -= psrc[1]; dz -= psrc[2];
      }
      FBl[j * 160 + mA * 10 + 0] = sqrtf(dx * dx + dy * dy + dz * dz);
    }

    // ---- A chunks for layer 1 (K = 19 padded to 20 -> 5 chunks), 2 tiles ----
    v2f Aa[5], Ab[5];
    Aa[0] = *(const v2f*)(SC0 + mA * 12 + 0 + mh * 2);
    Ab[0] = *(const v2f*)(SC1 + mA * 12 + 0 + mh * 2);
    Aa[1] = *(const v2f*)(SC0 + mA * 12 + 4 + mh * 2);
    Ab[1] = *(const v2f*)(SC1 + mA * 12 + 4 + mh * 2);
    Aa[2] = *(const v2f*)(SC0 + mA * 12 + 8 + mh * 2);
    Ab[2] = *(const v2f*)(SC1 + mA * 12 + 8 + mh * 2);
    Aa[3] = *(const v2f*)(FB0 + j * 160 + mA * 10 + 0 + mh * 2);
    Ab[3] = *(const v2f*)(FB1 + j * 160 + mA * 10 + 0 + mh * 2);
    Aa[4] = *(const v2f*)(FB0 + j * 160 + mA * 10 + 4 + mh * 2);
    Ab[4] = *(const v2f*)(FB1 + j * 160 + mA * 10 + 4 + mh * 2);

    // ---- layer 1: h = relu(in_feat @ W1^T + b1), N = 19 -> 2 N-tiles ----
    v8f c0a, c1a, c0b, c1b;
    const int   n1  = n + 16;
    const int   rr1 = (n1 < 19) ? n1 : 0;
    const float s1  = (n1 < 19) ? 1.f : 0.f;
    {
      const float bv0 = b1p[j * 19 + n];
      const float bv1 = (n1 < 19) ? b1p[j * 19 + n1] : 0.f;
      #pragma unroll
      for (int r = 0; r < 8; ++r) {
        c0a[r] = bv0; c0b[r] = bv0; c1a[r] = bv1; c1b[r] = bv1;
      }
    }
    #pragma unroll
    for (int t = 0; t < 5; ++t) {
      const int k0 = t * 4 + mh * 2;               // k0 <= 18
      v2f b0t, b1t;                                 // shared across both M-tiles
      b0t.x = w1j[n * 19 + k0];
      b0t.y = (k0 + 1 < 19) ? w1j[n * 19 + k0 + 1] : 0.f;
      b1t.x = w1j[rr1 * 19 + k0] * s1;
      b1t.y = (k0 + 1 < 19) ? w1j[rr1 * 19 + k0 + 1] * s1 : 0.f;
      c0a = wmma_f32(Aa[t], b0t, c0a);
      c0b = wmma_f32(Ab[t], b0t, c0b);
      c1a = wmma_f32(Aa[t], b1t, c1a);
      c1b = wmma_f32(Ab[t], b1t, c1b);
    }

    // ---- relu + transpose h to A-major LDS (slot 19 stays 0) ----
    #pragma unroll
    for (int r = 0; r < 8; ++r) {
      HB0[(r + 8 * mh) * 20 + n] = fmaxf(c0a[r], 0.f);
      HB1[(r + 8 * mh) * 20 + n] = fmaxf(c0b[r], 0.f);
    }
    if (n < 3) {
      #pragma unroll
      for (int r = 0; r < 8; ++r) {
        HB0[(r + 8 * mh) * 20 + 16 + n] = fmaxf(c1a[r], 0.f);
        HB1[(r + 8 * mh) * 20 + 16 + n] = fmaxf(c1b[r], 0.f);
      }
    }

    // ---- layer 2: out = h @ W2^T + b2, N = 6, K = 19 padded to 20 ----
    v8f c2a, c2b;
    {
      const float bv2 = (n < 6) ? b2p[j * 6 + n] : 0.f;
      #pragma unroll
      for (int r = 0; r < 8; ++r) { c2a[r] = bv2; c2b[r] = bv2; }
    }
    const int rr2 = (n < 6) ? n : 0;
    #pragma unroll
    for (int t = 0; t < 5; ++t) {
      const int k0 = t * 4 + mh * 2;
      v2f bt;
      bt.x = w2j[rr2 * 19 + k0] * nsel6;
      bt.y = (k0 + 1 < 19) ? w2j[rr2 * 19 + k0 + 1] * nsel6 : 0.f;
      v2f aa = *(const v2f*)(HB0 + mA * 20 + t * 4 + mh * 2);
      v2f ab = *(const v2f*)(HB1 + mA * 20 + t * 4 + mh * 2);
      c2a = wmma_f32(aa, bt, c2a);
      c2b = wmma_f32(ab, bt, c2b);
    }

    // ---- write out[j] and push feats into children's featbufs ----
    if (n < 6) {
      const size_t ob0 = (base + 8 * mh) * 144 + (size_t)j * 6 + n;
      const size_t ob1 = (base + 16 + 8 * mh) * 144 + (size_t)j * 6 + n;
      #pragma unroll
      for (int r = 0; r < 8; ++r) {
        outp[ob0 + (size_t)r * 144] = c2a[r];
        outp[ob1 + (size_t)r * 144] = c2b[r];
      }
      #pragma unroll
      for (int ci = 0; ci < 3; ++ci) {
        const int ch = c_child[j * 3 + ci];
        if (ch >= 0) {
          #pragma unroll
          for (int r = 0; r < 8; ++r) {
            FB0[ch * 160 + (r + 8 * mh) * 10 + 1 + n] = c2a[r];
            FB1[ch * 160 + (r + 8 * mh) * 10 + 1 + n] = c2b[r];
          }
        }
      }
    }
  }
}

extern "C" void kernel_launch(void* const* d_in, const int* in_sizes, int n_in,
                              void* d_out, int out_size, void* d_ws, size_t ws_size,
                              hipStream_t stream) {
  (void)n_in; (void)out_size; (void)d_ws; (void)ws_size;
  const float* rots = (const float*)d_in[0];
  const float* Jtrs = (const float*)d_in[1];
  const float* W0p  = (const float*)d_in[2];
  const float* b0p  = (const float*)d_in[3];
  const float* W1p  = (const float*)d_in[4];
  const float* b1p  = (const float*)d_in[5];
  const float* W2p  = (const float*)d_in[6];
  const float* b2p  = (const float*)d_in[7];
  float* outp = (float*)d_out;

  const int batch  = in_sizes[0] / (NJ * 9);   // 262144
  const int blocks = batch / 32;               // one wave (32 elems) per block

  hipLaunchKernelGGL(pose_encoder_kernel, dim3(blocks), dim3(THREADS), 0, stream,
                     rots, Jtrs, W0p, b0p, W1p, b1p, W2p, b2p, outp);
}